// FATMSparse_28870770164489
// MI455X (gfx1250) — compile-verified
//
#include <hip/hip_runtime.h>
#include <hip/hip_bf16.h>

// ---------------------------------------------------------------------------
// Shapes (from reference): T=4 B=4 C=256 NB=16 D=16 H=32 W=32
//   TB = 16, Hh = Wh = 16
// Layouts (row-major float32):
//   S    : (TB, 256, 32, 32)   spikes
//   L1   : (TB, 512, 32, 16)   HaarW + L1 threshold (pre-BN)
//   L2   : (TB, 1024, 16, 16)  band-gated quadrants LL|HL|LH|HH (pre-BN)
//   Hout : (TB, 1024, 16, 16)  block-diagonal matmul output
//   REC  : (TB, 256, 32, 32)   inverse Haar reconstruction (pre-BN)
//   c1   : staged in d_out, c2 in workspace (pre-BN)
// ---------------------------------------------------------------------------

typedef __attribute__((ext_vector_type(2))) float v2f;
typedef __attribute__((ext_vector_type(8))) float v8f;

#define S2F 0.70710678118654752f

__device__ __forceinline__ float bn_apply(float v, const float* __restrict__ st,
                                          const float* __restrict__ g,
                                          const float* __restrict__ b,
                                          int ch, float invN) {
  float mean = st[2 * ch] * invN;
  float var  = st[2 * ch + 1] * invN - mean * mean;
  float sc   = g[ch] * rsqrtf(var + 1e-5f);
  return (v - mean) * sc + b[ch];
}

// ---------------- LIF over T=4 (sequential scan per site) ------------------
__global__ void lif_kernel(const float* __restrict__ x, float* __restrict__ s) {
  int i = blockIdx.x * 256 + threadIdx.x;            // B*C*H*W = 1,048,576
  const int BCHW = 4 * 256 * 1024;
  float v = 0.f;
#pragma unroll
  for (int t = 0; t < 4; ++t) {
    float xt = x[t * BCHW + i];
    v = 0.5f * (v + xt);                              // v += (x - v)/tau, tau=2
    float sp = (v - 1.0f) >= 0.f ? 1.f : 0.f;        // spike(v - vth)
    s[t * BCHW + i] = sp;
    v *= (1.f - sp);
  }
}

// ------------- Haar along W (butterfly) + level-1 threshold ----------------
__global__ void haarw_kernel(const float* __restrict__ s, float* __restrict__ l1) {
  int i = blockIdx.x * 256 + threadIdx.x;            // 16*512*32*16 = 4,194,304
  int wq = i & 15, h = (i >> 4) & 31, c2 = (i >> 9) & 511, tt = i >> 18;
  int cc = c2 & 255;
  int base = ((tt * 256 + cc) * 32 + h) * 32 + 2 * wq;
  float a = s[base], b = s[base + 1];
  float v = (c2 >= 256) ? S2F * (a - b) : S2F * (a + b);
  v = (fabsf(v) >= 0.5f) ? v : 0.f;                  // v * spike(|v| - 0.5)
  l1[i] = v;
}

// ----- deterministic per-channel BN statistics (sum, sumsq), data (N,CH,S) --
__global__ void bn_stats_kernel(const float* __restrict__ data,
                                float* __restrict__ stats,
                                int N, int CH, int S) {
  int ch = blockIdx.x, tid = threadIdx.x;
  float s1 = 0.f, s2 = 0.f;
  int total = N * S;
  for (int idx = tid; idx < total; idx += 256) {
    int n = idx / S, sp = idx - n * S;
    float v = data[(n * CH + ch) * S + sp];
    s1 += v; s2 += v * v;
  }
  __shared__ float sh1[256], sh2[256];
  sh1[tid] = s1; sh2[tid] = s2; __syncthreads();
  for (int off = 128; off > 0; off >>= 1) {
    if (tid < off) { sh1[tid] += sh1[tid + off]; sh2[tid] += sh2[tid + off]; }
    __syncthreads();
  }
  if (tid == 0) { stats[2 * ch] = sh1[0]; stats[2 * ch + 1] = sh2[0]; }
}

// -------- bn_fwd + Haar along H + band_gate (per (tt,cc) slice) ------------
__global__ void haarh_bandgate_kernel(const float* __restrict__ l1,
                                      const float* __restrict__ st_fwd,
                                      const float* __restrict__ g,
                                      const float* __restrict__ b,
                                      float* __restrict__ l2) {
  int blk = blockIdx.x;                               // 16*256 = 4096
  int cc = blk & 255, tt = blk >> 8;
  int tid = threadIdx.x;                              // hq*16+wq
  int wq = tid & 15, hq = tid >> 4;
  const float invN = 1.f / 8192.f;                    // TB*H*Wh per l1 channel
  int cl = cc, ch = 256 + cc;
  float lo0 = bn_apply(l1[((tt * 512 + cl) * 32 + 2 * hq) * 16 + wq],     st_fwd, g, b, cl, invN);
  float lo1 = bn_apply(l1[((tt * 512 + cl) * 32 + 2 * hq + 1) * 16 + wq], st_fwd, g, b, cl, invN);
  float hi0 = bn_apply(l1[((tt * 512 + ch) * 32 + 2 * hq) * 16 + wq],     st_fwd, g, b, ch, invN);
  float hi1 = bn_apply(l1[((tt * 512 + ch) * 32 + 2 * hq + 1) * 16 + wq], st_fwd, g, b, ch, invN);
  float vals[4];
  vals[0] = S2F * (lo0 + lo1);  // LL
  vals[1] = S2F * (lo0 - lo1);  // HL
  vals[2] = S2F * (hi0 + hi1);  // LH
  vals[3] = S2F * (hi0 - hi1);  // HH
  const float taus[4] = {0.01f, 0.02f, 0.02f, 0.05f};
  __shared__ float sh[256];
#pragma unroll
  for (int q = 0; q < 4; ++q) {
    float c = vals[q];
    c = (fabsf(c) >= 0.5f) ? c : 0.f;                 // c * spike(|c| - 0.5)
    sh[tid] = c * c; __syncthreads();
    for (int off = 128; off > 0; off >>= 1) {
      if (tid < off) sh[tid] += sh[tid + off];
      __syncthreads();
    }
    float E = sh[0] * (1.f / 256.f);
    __syncthreads();
    float out = (E > taus[q]) ? c : 0.f;
    l2[((tt * 1024 + q * 256 + cc) * 16 + hq) * 16 + wq] = out;
  }
}

// ---- bn_mul + block-diagonal matmul via V_WMMA_F32_16X16X4_F32 ------------
// Per wave: one 16(pos) x 16(k) tile for (part, nb); K=16 in 4 WMMA K-steps.
__global__ void blockmm_kernel(const float* __restrict__ l2,
                               const float* __restrict__ st_mul,
                               const float* __restrict__ g,
                               const float* __restrict__ b,
                               const float* __restrict__ Wm,  // (NB,16,16)
                               float* __restrict__ hout) {
  int wave = blockIdx.x * (blockDim.x >> 5) + (threadIdx.x >> 5);  // 16384
  int lane = threadIdx.x & 31;
  int tile = wave & 255, nb = (wave >> 8) & 15, part = wave >> 12;
  int pos_base = tile * 16;
  int row = lane & 15, half = lane >> 4;
  int p = pos_base + row;
  int tt = p >> 8, hq = (p >> 4) & 15, wq = p & 15;
  const float invN = 1.f / 4096.f;                    // TB*Hh*Wh per l2 channel
  v8f acc = {};
#pragma unroll
  for (int kk = 0; kk < 4; ++kk) {
    int d0 = kk * 4 + 2 * half;
    int ch0 = nb * 64 + part * 16 + d0;
    v2f A, B;
    A.x = bn_apply(l2[((tt * 1024 + ch0) * 16 + hq) * 16 + wq],     st_mul, g, b, ch0, invN);
    A.y = bn_apply(l2[((tt * 1024 + ch0 + 1) * 16 + hq) * 16 + wq], st_mul, g, b, ch0 + 1, invN);
    B.x = Wm[(nb * 16 + d0) * 16 + row];
    B.y = Wm[(nb * 16 + d0 + 1) * 16 + row];
    acc = __builtin_amdgcn_wmma_f32_16x16x4_f32(false, A, false, B,
                                                (short)0, acc, false, false);
  }
#pragma unroll
  for (int r = 0; r < 8; ++r) {
    int m = r + 8 * half, n = row;
    int pp = pos_base + m;
    int t2 = pp >> 8, h2 = (pp >> 4) & 15, w2 = pp & 15;
    int cOut = nb * 64 + part * 16 + n;
    hout[((t2 * 1024 + cOut) * 16 + h2) * 16 + w2] = acc[r];
  }
}

// ------------------ inverse Haar (H then W), fully local --------------------
__global__ void invhaar_kernel(const float* __restrict__ hb, float* __restrict__ rec) {
  int i = blockIdx.x * 256 + threadIdx.x;             // 16*256*32*32
  int w = i & 31, h = (i >> 5) & 31, c = (i >> 10) & 255, tt = i >> 18;
  int hq = h >> 1, wq = w >> 1;
  float sh = (h & 1) ? -1.f : 1.f;
  float sw = (w & 1) ? -1.f : 1.f;
  int base = ((tt * 1024 + c) * 16 + hq) * 16 + wq;
  float LL = hb[base];
  float HL = hb[base + 256 * 256];
  float LH = hb[base + 512 * 256];
  float HH = hb[base + 768 * 256];
  rec[i] = 0.5f * (LL + sh * HL + sw * (LH + sh * HH));
}

// ------------- 1x1 conv (per (tt,nb) sample) via WMMA f32 ------------------
__global__ void conv1_kernel(const float* __restrict__ s,
                             const float* __restrict__ w1,   // (16,16,1,1)
                             const float* __restrict__ b1,
                             float* __restrict__ c1) {
  int wave = blockIdx.x * (blockDim.x >> 5) + (threadIdx.x >> 5);  // 16384
  int lane = threadIdx.x & 31;
  int tile = wave & 63, samp = wave >> 6;
  int tt = samp >> 4, nb = samp & 15;
  int pos_base = tile * 16;
  int row = lane & 15, half = lane >> 4;
  int p = pos_base + row;
  v8f acc = {};
#pragma unroll
  for (int kk = 0; kk < 4; ++kk) {
    int i0 = kk * 4 + 2 * half;
    v2f A, B;
    A.x = s[(tt * 256 + nb * 16 + i0) * 1024 + p];
    A.y = s[(tt * 256 + nb * 16 + i0 + 1) * 1024 + p];
    B.x = w1[row * 16 + i0];        // B[K=i][N=o] = w1[o][i]
    B.y = w1[row * 16 + i0 + 1];
    acc = __builtin_amdgcn_wmma_f32_16x16x4_f32(false, A, false, B,
                                                (short)0, acc, false, false);
  }
#pragma unroll
  for (int r = 0; r < 8; ++r) {
    int m = r + 8 * half, n = row;
    c1[(tt * 256 + nb * 16 + n) * 1024 + pos_base + m] = acc[r] + b1[n];
  }
}

// -------- 3x3 SAME conv as 9 shifted accumulated WMMA GEMMs ----------------
__global__ void conv2_kernel(const float* __restrict__ s,
                             const float* __restrict__ w2,   // (16,16,3,3)
                             const float* __restrict__ b2,
                             float* __restrict__ c2) {
  int wave = blockIdx.x * (blockDim.x >> 5) + (threadIdx.x >> 5);  // 16384
  int lane = threadIdx.x & 31;
  int tile = wave & 63, samp = wave >> 6;
  int tt = samp >> 4, nb = samp & 15;
  int pos_base = tile * 16;
  int row = lane & 15, half = lane >> 4;
  int p = pos_base + row;
  int hh = p >> 5, ww = p & 31;
  v8f acc = {};
#pragma unroll
  for (int dy = 0; dy < 3; ++dy) {
#pragma unroll
    for (int dx = 0; dx < 3; ++dx) {
      int h2 = hh + dy - 1, w2i = ww + dx - 1;
      bool ok = (h2 >= 0) && (h2 < 32) && (w2i >= 0) && (w2i < 32);
      int sp = h2 * 32 + w2i;
#pragma unroll
      for (int kk = 0; kk < 4; ++kk) {
        int i0 = kk * 4 + 2 * half;
        v2f A, B;
        A.x = ok ? s[(tt * 256 + nb * 16 + i0) * 1024 + sp] : 0.f;
        A.y = ok ? s[(tt * 256 + nb * 16 + i0 + 1) * 1024 + sp] : 0.f;
        B.x = w2[((row * 16 + i0) * 3 + dy) * 3 + dx];
        B.y = w2[((row * 16 + i0 + 1) * 3 + dy) * 3 + dx];
        acc = __builtin_amdgcn_wmma_f32_16x16x4_f32(false, A, false, B,
                                                    (short)0, acc, false, false);
      }
    }
  }
#pragma unroll
  for (int r = 0; r < 8; ++r) {
    int m = r + 8 * half, n = row;
    c2[(tt * 256 + nb * 16 + n) * 1024 + pos_base + m] = acc[r] + b2[n];
  }
}

// ---------------- final: bn_inv(rec) + bn_c1(c1) + bn_c2(c2) ---------------
__global__ void final_kernel(const float* __restrict__ rec,
                             const float* __restrict__ c2,
                             const float* __restrict__ st_inv,
                             const float* __restrict__ st_c1,
                             const float* __restrict__ st_c2,
                             const float* __restrict__ gi, const float* __restrict__ bi,
                             const float* __restrict__ g1, const float* __restrict__ b1,
                             const float* __restrict__ g2, const float* __restrict__ b2,
                             float* __restrict__ out) {
  int i = blockIdx.x * 256 + threadIdx.x;             // 4,194,304
  int c = (i >> 10) & 255;
  const float invN = 1.f / 16384.f;                   // TB*H*W per channel
  float r = bn_apply(rec[i], st_inv, gi, bi, c, invN);
  float a = bn_apply(out[i], st_c1, g1, b1, c, invN); // c1 staged in d_out
  float d = bn_apply(c2[i],  st_c2, g2, b2, c, invN);
  out[i] = r + a + d;
}

// ---------------------------------------------------------------------------
extern "C" void kernel_launch(void* const* d_in, const int* in_sizes, int n_in,
                              void* d_out, int out_size, void* d_ws, size_t ws_size,
                              hipStream_t stream) {
  const float* x   = (const float*)d_in[0];
  const float* hw  = (const float*)d_in[1];   // haar_weight (16,16,16)
  const float* w1  = (const float*)d_in[2];
  const float* b1  = (const float*)d_in[3];
  const float* w2  = (const float*)d_in[4];
  const float* b2  = (const float*)d_in[5];
  const float* gF  = (const float*)d_in[6];
  const float* bF  = (const float*)d_in[7];
  const float* gM  = (const float*)d_in[8];
  const float* bM  = (const float*)d_in[9];
  const float* gI  = (const float*)d_in[10];
  const float* bI  = (const float*)d_in[11];
  const float* g1  = (const float*)d_in[12];
  const float* b1g = (const float*)d_in[13];
  const float* g2  = (const float*)d_in[14];
  const float* b2g = (const float*)d_in[15];

  const size_t BUF = 4194304;                  // floats per big buffer
  float* ws   = (float*)d_ws;
  float* S    = ws;                            // spikes
  float* bufA = ws + BUF;                      // L1 raw -> Hout -> c2
  float* bufB = ws + 2 * BUF;                  // L2 -> rec
  float* st   = ws + 3 * BUF;                  // stats
  float* stF  = st;                            // 512 ch * 2
  float* stM  = st + 1024;                     // 1024 ch * 2
  float* stI  = st + 3072;                     // 256 ch * 2
  float* stC1 = st + 3584;                     // 256 ch * 2
  float* stC2 = st + 4096;                     // 256 ch * 2
  float* outF = (float*)d_out;

  lif_kernel<<<4096, 256, 0, stream>>>(x, S);
  haarw_kernel<<<16384, 256, 0, stream>>>(S, bufA);
  bn_stats_kernel<<<512, 256, 0, stream>>>(bufA, stF, 16, 512, 512);
  haarh_bandgate_kernel<<<4096, 256, 0, stream>>>(bufA, stF, gF, bF, bufB);
  bn_stats_kernel<<<1024, 256, 0, stream>>>(bufB, stM, 16, 1024, 256);
  blockmm_kernel<<<2048, 256, 0, stream>>>(bufB, stM, gM, bM, hw, bufA);
  invhaar_kernel<<<16384, 256, 0, stream>>>(bufA, bufB);      // rec in bufB
  bn_stats_kernel<<<256, 256, 0, stream>>>(bufB, stI, 16, 256, 1024);
  conv1_kernel<<<2048, 256, 0, stream>>>(S, w1, b1, outF);    // c1 in d_out
  bn_stats_kernel<<<256, 256, 0, stream>>>(outF, stC1, 16, 256, 1024);
  conv2_kernel<<<2048, 256, 0, stream>>>(S, w2, b2, bufA);    // c2 in bufA
  bn_stats_kernel<<<256, 256, 0, stream>>>(bufA, stC2, 16, 256, 1024);
  final_kernel<<<16384, 256, 0, stream>>>(bufB, bufA, stI, stC1, stC2,
                                          gI, bI, g1, b1g, g2, b2g, outF);
  (void)in_sizes; (void)n_in; (void)out_size; (void)ws_size;
}